// FullTransportRouting_12463995093503
// MI455X (gfx1250) — compile-verified
//
#include <hip/hip_runtime.h>
#include <hip/hip_bf16.h>
#include <math.h>

typedef float v2f __attribute__((ext_vector_type(2)));
typedef float v8f __attribute__((ext_vector_type(8)));

#define TOK   8192      // B*T
#define SDIM  4096      // M*C
#define MM    16
#define CC    256
#define KK    4
#define DD    64
#define LDIM  1024
#define NIT   20
#define KSPL  8         // K-split factor for the code GEMM
#define KCH   (SDIM / KSPL)   // 512

// ---------------- kernel 0: Cg = diag(gamma+1) @ compress ----------------
__global__ void k0_prep_cg(const float* __restrict__ gamma,
                           const float* __restrict__ compress,
                           float* __restrict__ cg) {
    int i = blockIdx.x * blockDim.x + threadIdx.x;        // over 4096*64
    if (i < SDIM * DD) {
        int krow = i >> 6;                                // which of 4096
        cg[i] = compress[i] * (gamma[krow] + 1.0f);
    }
}

// ---------------- kernel 1: per-token inverse L2 (times sqrt(4096)) ------
__global__ void k1_norm(const float* __restrict__ s, float* __restrict__ inv) {
    __shared__ float red[256];
    int tok = blockIdx.x;
    const float* sp = s + (size_t)tok * SDIM;
    float sum = 0.f;
    for (int i = threadIdx.x; i < SDIM; i += 256) {
        float v = sp[i];
        sum += v * v;
    }
    red[threadIdx.x] = sum;
    __syncthreads();
    for (int off = 128; off > 0; off >>= 1) {
        if (threadIdx.x < off) red[threadIdx.x] += red[threadIdx.x + off];
        __syncthreads();
    }
    if (threadIdx.x == 0)
        inv[tok] = 64.0f / fmaxf(sqrtf(red[0]), 1e-12f);  // 64 = sqrt(4096)
}

// ---------------- kernel 2: partial code GEMM (WMMA f32 16x16x4) ---------
// grid: (128, 8) x 128 threads (4 waves); wave = one 16-token tile,
// one K-chunk of 512, full N=64. Partials go to w_cpart[ks][tok][n].
__global__ void k2_code_part(const float* __restrict__ s,
                             const float* __restrict__ cg,
                             float* __restrict__ cpart) {
    int wave = threadIdx.x >> 5;
    int lane = threadIdx.x & 31;
    int tt   = blockIdx.x * 4 + wave;            // 0..511 token tiles
    int ks   = blockIdx.y;                       // 0..7 K chunk
    int mrow = lane & 15;
    int sub  = (lane >> 4) * 2;                  // K sub-offset 0 or 2
    const float* arow = s + (size_t)(tt * 16 + mrow) * SDIM + ks * KCH + sub;
    const float* bbase = cg + (size_t)ks * KCH * DD;
    v8f acc0 = {}, acc1 = {}, acc2 = {}, acc3 = {};
#pragma unroll 1
    for (int kk = 0; kk < KCH / 4; ++kk) {
        int k4 = kk * 4;
        v2f a = *(const v2f*)(arow + k4);
        const float* b0 = bbase + (size_t)(k4 + sub) * DD + mrow;
        v2f b;
        b.x = b0[0];      b.y = b0[DD];
        acc0 = __builtin_amdgcn_wmma_f32_16x16x4_f32(false, a, false, b, (short)0, acc0, false, false);
        b.x = b0[16];     b.y = b0[DD + 16];
        acc1 = __builtin_amdgcn_wmma_f32_16x16x4_f32(false, a, false, b, (short)0, acc1, false, false);
        b.x = b0[32];     b.y = b0[DD + 32];
        acc2 = __builtin_amdgcn_wmma_f32_16x16x4_f32(false, a, false, b, (short)0, acc2, false, false);
        b.x = b0[48];     b.y = b0[DD + 48];
        acc3 = __builtin_amdgcn_wmma_f32_16x16x4_f32(false, a, false, b, (short)0, acc3, false, false);
    }
    // C layout: vgpr r -> M = r + (lane>>4)*8, N = lane&15
    int halo = (lane >> 4) * 8;
    int n    = lane & 15;
#pragma unroll
    for (int r = 0; r < 8; ++r) {
        int tok = tt * 16 + r + halo;
        float* crow = cpart + ((size_t)ks * TOK + tok) * DD + n;
        crow[0]  = acc0[r];
        crow[16] = acc1[r];
        crow[32] = acc2[r];
        crow[48] = acc3[r];
    }
}

// ---------------- kernel 2r: reduce K-split partials, apply inv scale ----
__global__ void k2_code_reduce(const float* __restrict__ cpart,
                               const float* __restrict__ inv,
                               float* __restrict__ code) {
    int i = blockIdx.x * blockDim.x + threadIdx.x;   // over 8192*64
    if (i < TOK * DD) {
        int tok = i >> 6;
        float acc = 0.f;
#pragma unroll
        for (int ks = 0; ks < KSPL; ++ks)
            acc += cpart[(size_t)ks * TOK * DD + i];
        code[i] = acc * inv[tok];
    }
}

// ---------------- kernel 3: logits + sinkhorn + layer_in -----------------
// grid: 8192 blocks x 64 threads; group g = tid>>4 handles one route matrix
__global__ void k3_sinkhorn(const float* __restrict__ s,
                            const float* __restrict__ code,
                            const float* __restrict__ a_pre,  const float* __restrict__ tl_pre,  const float* __restrict__ h_pre,
                            const float* __restrict__ a_align,const float* __restrict__ tl_align,const float* __restrict__ h_align,
                            const float* __restrict__ a_next, const float* __restrict__ tl_next, const float* __restrict__ h_next,
                            const float* __restrict__ a_post, const float* __restrict__ tl_post, const float* __restrict__ h_post,
                            float* __restrict__ p_align, float* __restrict__ p_next,
                            float* __restrict__ p_post,  float* __restrict__ lin) {
    __shared__ float cde[DD];
    __shared__ float Z[4][16][16];
    __shared__ float Tt[4][16][16];
    __shared__ float vv[4][16];
    __shared__ float Ppre[KK * 16];

    int tok = blockIdx.x;
    int tid = threadIdx.x;
    int g   = tid >> 4;
    int r   = tid & 15;

    if (tid < DD) cde[tid] = code[(size_t)tok * DD + tid];
    ((float*)vv)[tid] = 0.0f;
    __syncthreads();

    int rows = (g == 0) ? KK : 16;
    int cols = (g == 3) ? KK : 16;
    float alpha = (g == 0) ? a_pre[0] : (g == 1) ? a_align[0] : (g == 2) ? a_next[0] : a_post[0];
    const float* tl   = (g == 0) ? tl_pre : (g == 1) ? tl_align : (g == 2) ? tl_next : tl_post;
    const float* bias = (g == 0) ? h_pre  : (g == 1) ? h_align  : (g == 2) ? h_next  : h_post;

    if (r < rows) {
        int rc = rows * cols;
        for (int c = 0; c < cols; ++c) {
            float dot = 0.f;
#pragma unroll 4
            for (int d = 0; d < DD; ++d)
                dot += cde[d] * tl[d * rc + r * cols + c];
            Z[g][r][c] = alpha * dot + bias[r * cols + c];   // tau = 1
        }
    }
    __syncthreads();

    float lr = -logf((float)rows);
    float lc = -logf((float)cols);
    float u  = 0.f;
    for (int it = 0; it < NIT; ++it) {
        if (r < rows) {
            float mx = -1e30f;
            for (int c = 0; c < cols; ++c) mx = fmaxf(mx, Z[g][r][c] + vv[g][c]);
            float ss = 0.f;
            for (int c = 0; c < cols; ++c) ss += expf(Z[g][r][c] + vv[g][c] - mx);
            u = lr - (mx + logf(ss));
            for (int c = 0; c < cols; ++c) Tt[g][c][r] = Z[g][r][c] + u;
        }
        __syncthreads();
        if (r < cols) {
            float mx = -1e30f;
            for (int rr = 0; rr < rows; ++rr) mx = fmaxf(mx, Tt[g][r][rr]);
            float ss = 0.f;
            for (int rr = 0; rr < rows; ++rr) ss += expf(Tt[g][r][rr] - mx);
            vv[g][r] = lc - (mx + logf(ss));
        }
        __syncthreads();
    }

    if (r < rows) {
        for (int c = 0; c < cols; ++c) {
            float val = expf(Z[g][r][c] + u + vv[g][c]) * (float)rows;
            if (g == 0)      Ppre[r * 16 + c] = val;
            else if (g == 1) p_align[(size_t)tok * 256 + r * 16 + c] = val;
            else if (g == 2) p_next [(size_t)tok * 256 + r * 16 + c] = val;
            else             p_post [(size_t)tok * 64  + r * KK + c] = val;
        }
    }
    __syncthreads();

    // layer_in = P_pre (4x16) @ s (16x256)
    const float* sp = s + (size_t)tok * SDIM;
    for (int idx = tid; idx < KK * CC; idx += 64) {
        int kk = idx >> 8;
        int cc = idx & 255;
        float acc = 0.f;
#pragma unroll
        for (int m = 0; m < MM; ++m)
            acc += Ppre[kk * 16 + m] * sp[m * CC + cc];
        lin[(size_t)tok * LDIM + idx] = acc;
    }
}

// ---------------- kernel 4: layer_out = layer_in @ W  (WMMA f32) ---------
// grid: (512, 4) x 128 threads (4 waves); wave handles 16 tokens x 64 cols
__global__ void k4_wgemm(const float* __restrict__ lin,
                         const float* __restrict__ W,
                         float* __restrict__ lo) {
    int wave = threadIdx.x >> 5;
    int lane = threadIdx.x & 31;
    int tt   = blockIdx.x;                       // token tile 0..511
    int n64  = blockIdx.y * 4 + wave;            // 0..15
    int mrow = lane & 15;
    int sub  = (lane >> 4) * 2;
    const float* arow = lin + (size_t)(tt * 16 + mrow) * LDIM + sub;
    v8f acc0 = {}, acc1 = {}, acc2 = {}, acc3 = {};
#pragma unroll 1
    for (int kk = 0; kk < LDIM / 4; ++kk) {
        int k4 = kk * 4;
        v2f a = *(const v2f*)(arow + k4);
        const float* b0 = W + (size_t)(k4 + sub) * LDIM + n64 * 64 + mrow;
        v2f b;
        b.x = b0[0];  b.y = b0[LDIM];
        acc0 = __builtin_amdgcn_wmma_f32_16x16x4_f32(false, a, false, b, (short)0, acc0, false, false);
        b.x = b0[16]; b.y = b0[LDIM + 16];
        acc1 = __builtin_amdgcn_wmma_f32_16x16x4_f32(false, a, false, b, (short)0, acc1, false, false);
        b.x = b0[32]; b.y = b0[LDIM + 32];
        acc2 = __builtin_amdgcn_wmma_f32_16x16x4_f32(false, a, false, b, (short)0, acc2, false, false);
        b.x = b0[48]; b.y = b0[LDIM + 48];
        acc3 = __builtin_amdgcn_wmma_f32_16x16x4_f32(false, a, false, b, (short)0, acc3, false, false);
    }
    int halo = (lane >> 4) * 8;
    int n    = lane & 15;
#pragma unroll
    for (int r = 0; r < 8; ++r) {
        int tok = tt * 16 + r + halo;
        float* crow = lo + (size_t)tok * LDIM + n64 * 64 + n;
        crow[0]  = acc0[r];
        crow[16] = acc1[r];
        crow[32] = acc2[r];
        crow[48] = acc3[r];
    }
}

// ---------------- kernel 5: s_next = P_next @ (P_align@s + P_post@out) ---
// grid: 8192 blocks x 256 threads; thread owns one column c of the token
__global__ void k5_finalize(const float* __restrict__ s,
                            const float* __restrict__ lo,
                            const float* __restrict__ p_align,
                            const float* __restrict__ p_next,
                            const float* __restrict__ p_post,
                            float* __restrict__ out) {
    __shared__ float Pa[256], Pn[256], Pp[64];
    int tok = blockIdx.x;
    int c   = threadIdx.x;
    Pa[c] = p_align[(size_t)tok * 256 + c];
    Pn[c] = p_next [(size_t)tok * 256 + c];
    if (c < 64) Pp[c] = p_post[(size_t)tok * 64 + c];
    __syncthreads();

    const float* sp = s  + (size_t)tok * SDIM + c;
    const float* op = lo + (size_t)tok * LDIM + c;
    float sv[MM], ov[KK], su[MM];
#pragma unroll
    for (int m = 0; m < MM; ++m) sv[m] = sp[m * CC];
#pragma unroll
    for (int k = 0; k < KK; ++k) ov[k] = op[k * CC];
#pragma unroll
    for (int m = 0; m < MM; ++m) {
        float acc = 0.f;
#pragma unroll
        for (int j = 0; j < MM; ++j) acc += Pa[m * 16 + j] * sv[j];
#pragma unroll
        for (int k = 0; k < KK; ++k) acc += Pp[m * KK + k] * ov[k];
        su[m] = acc;
    }
    float* po = out + (size_t)tok * SDIM + c;
#pragma unroll
    for (int m = 0; m < MM; ++m) {
        float acc = 0.f;
#pragma unroll
        for (int j = 0; j < MM; ++j) acc += Pn[m * 16 + j] * su[j];
        po[m * CC] = acc;
    }
}

// -------------------------------------------------------------------------
extern "C" void kernel_launch(void* const* d_in, const int* in_sizes, int n_in,
                              void* d_out, int out_size, void* d_ws, size_t ws_size,
                              hipStream_t stream) {
    const float* s        = (const float*)d_in[0];
    const float* gamma    = (const float*)d_in[1];
    const float* compress = (const float*)d_in[2];
    const float* a_pre    = (const float*)d_in[3];
    const float* tl_pre   = (const float*)d_in[4];
    const float* h_pre    = (const float*)d_in[5];
    const float* a_align  = (const float*)d_in[6];
    const float* tl_align = (const float*)d_in[7];
    const float* h_align  = (const float*)d_in[8];
    const float* a_next   = (const float*)d_in[9];
    const float* tl_next  = (const float*)d_in[10];
    const float* h_next   = (const float*)d_in[11];
    const float* a_post   = (const float*)d_in[12];
    const float* tl_post  = (const float*)d_in[13];
    const float* h_post   = (const float*)d_in[14];
    const float* W        = (const float*)d_in[15];
    float* outp = (float*)d_out;

    float* w_inv   = (float*)d_ws;                     // 8192
    float* w_cg    = w_inv   + TOK;                    // 4096*64
    float* w_cpart = w_cg    + SDIM * DD;              // 8 * 8192*64
    float* w_code  = w_cpart + (size_t)KSPL * TOK * DD;// 8192*64
    float* w_pa    = w_code  + TOK * DD;               // 8192*256
    float* w_pn    = w_pa    + TOK * 256;              // 8192*256
    float* w_pp    = w_pn    + TOK * 256;              // 8192*64
    float* w_lin   = w_pp    + TOK * 64;               // 8192*1024
    float* w_lo    = w_lin   + (size_t)TOK * LDIM;     // 8192*1024

    k0_prep_cg<<<(SDIM * DD + 255) / 256, 256, 0, stream>>>(gamma, compress, w_cg);
    k1_norm<<<TOK, 256, 0, stream>>>(s, w_inv);
    k2_code_part<<<dim3(TOK / 16 / 4, KSPL), 128, 0, stream>>>(s, w_cg, w_cpart);
    k2_code_reduce<<<(TOK * DD + 255) / 256, 256, 0, stream>>>(w_cpart, w_inv, w_code);
    k3_sinkhorn<<<TOK, 64, 0, stream>>>(s, w_code,
                                        a_pre, tl_pre, h_pre,
                                        a_align, tl_align, h_align,
                                        a_next, tl_next, h_next,
                                        a_post, tl_post, h_post,
                                        w_pa, w_pn, w_pp, w_lin);
    k4_wgemm<<<dim3(TOK / 16, 4), 128, 0, stream>>>(w_lin, W, w_lo);
    k5_finalize<<<TOK, 256, 0, stream>>>(s, w_lo, w_pa, w_pn, w_pp, outp);
}